// GlobalHypergraphPooling_67010079752618
// MI455X (gfx1250) — compile-verified
//
#include <hip/hip_runtime.h>
#include <hip/hip_bf16.h>

typedef __attribute__((ext_vector_type(2))) float v2f;
typedef __attribute__((ext_vector_type(8))) float v8f;

// One wave (32 lanes) per graph. batch[] is sorted, so graph g owns the
// contiguous row range [lower_bound(g), lower_bound(g+1)).
// Row-sum computed as D = A(ones 16x4) x B(4 rows x 16 ch) + C via
// V_WMMA_F32_16X16X4_F32 (exact f32 FMA accumulate). Every row of D holds the
// same channel sums; row M=0 is accumulator element 0 on lanes 0..15.
//
// B-matrix 4x16 lane mapping: N = lane&15; lanes 0-15 carry K=0,1 (rows r,r+1),
// lanes 16-31 carry K=2,3 (rows r+2,r+3).
__global__ __launch_bounds__(32)
void GlobalHypergraphPooling_segmean_wmma(const float* __restrict__ x,
                                          const int* __restrict__ batch,
                                          float* __restrict__ out,
                                          int n_rows) {
  const int g    = blockIdx.x;
  const int lane = threadIdx.x;  // wave32

  // Binary searches (uniform across the wave).
  int lo = 0, hi = n_rows;
  while (lo < hi) {
    int mid = (lo + hi) >> 1;
    if (batch[mid] < g) lo = mid + 1; else hi = mid;
  }
  const int start = lo;
  hi = n_rows;
  while (lo < hi) {
    int mid = (lo + hi) >> 1;
    if (batch[mid] < g + 1) lo = mid + 1; else hi = mid;
  }
  const int count = lo - start;

  // A-matrix 16x4 of ones (2 VGPRs/lane).
  v2f a;
  a.x = 1.0f;
  a.y = 1.0f;

  // 8 channel groups of 16 -> 128 channels; 8 accumulator VGPRs each.
  v8f acc[8] = {};

  const int col  = lane & 15;
  const int rofs = (lane >> 4) * 2;  // 0 for lanes 0-15, 2 for lanes 16-31

  // Per-lane marching pointer; all hot-loop loads use immediate offsets.
  const float* p = x + (size_t)(start + rofs) * 128 + col;

  int i = 0;
  const int c8 = count & ~7;

  // Main loop: 8 rows per iteration, branchless. 32 loads issued back-to-back,
  // then 16 WMMAs -> deep MLP for the bandwidth-bound stream.
  for (; i < c8; i += 8) {
    float b0x[8], b0y[8], b1x[8], b1y[8];
#pragma unroll
    for (int cg = 0; cg < 8; ++cg) {
      b0x[cg] = p[cg * 16];              // row r+rofs
      b0y[cg] = p[cg * 16 + 128];        // row r+rofs+1
      b1x[cg] = p[cg * 16 + 512];        // row r+4+rofs
      b1y[cg] = p[cg * 16 + 640];        // row r+4+rofs+1
    }
#pragma unroll
    for (int cg = 0; cg < 8; ++cg) {
      v2f b; b.x = b0x[cg]; b.y = b0y[cg];
      acc[cg] = __builtin_amdgcn_wmma_f32_16x16x4_f32(
          false, a, false, b, (short)0, acc[cg], false, false);
    }
#pragma unroll
    for (int cg = 0; cg < 8; ++cg) {
      v2f b; b.x = b1x[cg]; b.y = b1y[cg];
      acc[cg] = __builtin_amdgcn_wmma_f32_16x16x4_f32(
          false, a, false, b, (short)0, acc[cg], false, false);
    }
    p += 8 * 128;
  }

  // One unguarded 4-row step if available.
  if (i + 4 <= count) {
#pragma unroll
    for (int cg = 0; cg < 8; ++cg) {
      v2f b;
      b.x = p[cg * 16];
      b.y = p[cg * 16 + 128];
      acc[cg] = __builtin_amdgcn_wmma_f32_16x16x4_f32(
          false, a, false, b, (short)0, acc[cg], false, false);
    }
    i += 4;
  }

  // Branchless masked tail (1..3 rows): clamp row index into the valid range
  // and zero the contribution with a v_cndmask-style multiply.
  if (i < count) {
    const int rem = count - i;  // 1..3
    const float m0 = (rofs < rem) ? 1.0f : 0.0f;
    const float m1 = (rofs + 1 < rem) ? 1.0f : 0.0f;
    const int o0 = (rofs     < rem ? rofs     : rem - 1) * 128;
    const int o1 = (rofs + 1 < rem ? rofs + 1 : rem - 1) * 128;
    const float* q = x + (size_t)(start + i) * 128 + col;
#pragma unroll
    for (int cg = 0; cg < 8; ++cg) {
      v2f b;
      b.x = q[o0 + cg * 16] * m0;
      b.y = q[o1 + cg * 16] * m1;
      acc[cg] = __builtin_amdgcn_wmma_f32_16x16x4_f32(
          false, a, false, b, (short)0, acc[cg], false, false);
    }
  }

  // D row M=0: element 0 of the accumulator on lanes 0..15.
  const float inv = 1.0f / (float)(count > 0 ? count : 1);
  if (lane < 16) {
#pragma unroll
    for (int cg = 0; cg < 8; ++cg) {
      out[(size_t)g * 128 + cg * 16 + lane] = acc[cg][0] * inv;
    }
  }
}

extern "C" void kernel_launch(void* const* d_in, const int* in_sizes, int n_in,
                              void* d_out, int out_size, void* d_ws, size_t ws_size,
                              hipStream_t stream) {
  const float* x     = (const float*)d_in[0];
  const int*   batch = (const int*)d_in[1];
  float*       out   = (float*)d_out;

  const int n_rows     = in_sizes[1];            // 1,000,000
  const int channels   = in_sizes[0] / n_rows;   // 128
  const int num_graphs = out_size / channels;    // 4096

  (void)n_in; (void)d_ws; (void)ws_size;

  GlobalHypergraphPooling_segmean_wmma<<<num_graphs, 32, 0, stream>>>(
      x, batch, out, n_rows);
}